// Item_Graph_14620068675899
// MI455X (gfx1250) — compile-verified
//
#include <hip/hip_runtime.h>

// CDNA5 / gfx1250: wave32, WMMA 16x16x4 f32.
typedef float v2f __attribute__((ext_vector_type(2)));
typedef float v8f __attribute__((ext_vector_type(8)));

#define N_ITEMS 100000
#define LATDIM  128
#define KNN_K   5
#define TILES   (N_ITEMS / 16)          // 6250 16-row tiles
#define EDGE_W  0.2f                    // (5+1e-7)^-0.5 squared == 1/5 in f32

// One wave = one 16-row tile.
//   OUT(16x128) = S(16x80) x G(80x128),  S[i,g] = w iff g/5 == i (one-hot band),
//   G[g,:] = src[knn_flat[80*tile + g], :]   (80 contiguous indices per tile).
// K=80 split into 20 blocks of 4 -> V_WMMA_F32_16X16X4_F32 accumulation.
//
// VGPR layouts (ISA 7.12.2, 32-bit):
//   A 16x4 : lane L (M = L&15, h = L>>4), vgpr j holds K-elem (2h + j)
//   B 4x16 : lane L (N = L&15, h = L>>4), vgpr j holds K-row  (2h + j)
//   C 16x16: lane L (N = L&15, h = L>>4), vgpr v holds row    (v + 8h)
template <bool ADD_INPUTS>
__global__ __launch_bounds__(256) void spmm_gcn_wmma(
    const float* __restrict__ src,   // gather source [N,128]
    const int*   __restrict__ knn,   // [N,5] flat
    const float* __restrict__ x0,    // residual input (ADD_INPUTS)
    const float* __restrict__ x1,    // residual input (ADD_INPUTS)
    float*       __restrict__ dst)   // [N,128]
{
    const int wave = (int)((blockIdx.x * blockDim.x + threadIdx.x) >> 5);
    if (wave >= TILES) return;                 // wave-uniform guard: EXEC stays all-1s
    const int lane = threadIdx.x & 31;
    const int h    = lane >> 4;                // half-wave select
    const int c    = lane & 15;                // N column / A row (M)
    const int row0 = wave * 16;

    // ---- preload the tile's 80 contiguous neighbor indices into 3 VGPRs ----
    const int base = row0 * KNN_K;             // == 80 * wave
    const int i0 = knn[base + lane];
    const int i1 = knn[base + 32 + lane];
    int t2 = base + 64 + lane;                 // only g<80 ever consumed; clamp OOB lanes
    if (t2 > N_ITEMS * KNN_K - 1) t2 = N_ITEMS * KNN_K - 1;
    const int i2 = knn[t2];

    // ---- accumulators: 8 n-tiles of 16x16 f32 (64 VGPRs) ----
    v8f acc[8];
#pragma unroll
    for (int nt = 0; nt < 8; ++nt) {
        if (ADD_INPUTS) {
#pragma unroll
            for (int v = 0; v < 8; ++v) {
                const size_t e = (size_t)(row0 + v + 8 * h) * LATDIM + nt * 16 + c;
                acc[nt][v] = x0[e] + x1[e];    // fold residual adds into WMMA C init
            }
        } else {
#pragma unroll
            for (int v = 0; v < 8; ++v) acc[nt][v] = 0.0f;
        }
    }

    // ---- 20 K-blocks of 4 gathered rows each ----
#pragma unroll
    for (int kb = 0; kb < 20; ++kb) {
        const int g0 = 4 * kb + 2 * h;         // neighbor slot for this lane's B row j=0
        int ia, ib;                            // gathered row ids (ds_bpermute broadcast)
        if (kb < 8)       { ia = __shfl(i0, g0,      32); ib = __shfl(i0, g0 + 1,  32); }
        else if (kb < 16) { ia = __shfl(i1, g0 - 32, 32); ib = __shfl(i1, g0 - 31, 32); }
        else              { ia = __shfl(i2, g0 - 64, 32); ib = __shfl(i2, g0 - 63, 32); }

        // One-hot band A: S[m, g] = w iff g/5 == m   (m == c for this lane)
        v2f a;
        a.x = ((g0 / KNN_K) == c)       ? EDGE_W : 0.0f;
        a.y = (((g0 + 1) / KNN_K) == c) ? EDGE_W : 0.0f;

        const float* pa = src + (size_t)ia * LATDIM + c;   // 64B-coalesced row segments
        const float* pb = src + (size_t)ib * LATDIM + c;

#pragma unroll
        for (int nt = 0; nt < 8; ++nt) {
            v2f b;
            b.x = pa[nt * 16];
            b.y = pb[nt * 16];
            acc[nt] = __builtin_amdgcn_wmma_f32_16x16x4_f32(
                /*neg_a=*/false, a, /*neg_b=*/false, b,
                /*c_mod=*/(short)0, acc[nt],
                /*reuse_a=*/false, /*reuse_b=*/false);
        }
    }

    // ---- store D (row-major, 64B segments per half-wave) ----
#pragma unroll
    for (int nt = 0; nt < 8; ++nt) {
#pragma unroll
        for (int v = 0; v < 8; ++v) {
            dst[(size_t)(row0 + v + 8 * h) * LATDIM + nt * 16 + c] = acc[nt][v];
        }
    }
}

extern "C" void kernel_launch(void* const* d_in, const int* in_sizes, int n_in,
                              void* d_out, int out_size, void* d_ws, size_t ws_size,
                              hipStream_t stream) {
    (void)in_sizes; (void)n_in; (void)out_size; (void)ws_size;
    const float* item_rep = (const float*)d_in[0];
    const int*   knn      = (const int*)d_in[1];
    float*       out      = (float*)d_out;
    float*       x1       = (float*)d_ws;      // layer-1 result, 100000*128*4 = 51.2 MB

    dim3 blk(256);                              // 8 waves / block
    dim3 grd((TILES * 32 + 255) / 256);         // 782 blocks covering 6250 tile-waves

    // Layer 1: x1 = 0.2 * Gather5(item_rep)
    spmm_gcn_wmma<false><<<grd, blk, 0, stream>>>(item_rep, knn, nullptr, nullptr, x1);
    // Layer 2 + residuals: out = item_rep + x1 + 0.2 * Gather5(x1)
    spmm_gcn_wmma<true><<<grd, blk, 0, stream>>>(x1, knn, item_rep, x1, out);
}